// RNN_54898271977981
// MI455X (gfx1250) — compile-verified
//
#include <hip/hip_runtime.h>
#include <math.h>

typedef __attribute__((ext_vector_type(2))) float v2f;
typedef __attribute__((ext_vector_type(8))) float v8f;

#define Bsz 64
#define Tsz 2048
#define Hsz 256
#define KT  64      // K tiles of 4 (256 / 4)
#define HS  260     // padded LDS row stride in floats (260 % 64 == 4 -> conflict-free b64 reads)

// --------------------------------------------------------------------------
// pre_gemm: out[M,256] = X[M,256] @ W^T + (bi + bh)
//   (X@W^T)[m,n] = sum_k X[m,k] * W[n,k],  M = B*T = 131072
// One wave computes a 64x16 output slab (4 M-tiles x 1 N-tile), K=256 via
// 64 chained V_WMMA_F32_16X16X4_F32 per tile; B-fragment shared by 4 tiles.
// --------------------------------------------------------------------------
__global__ __launch_bounds__(256)
void pre_gemm(const float* __restrict__ X, const float* __restrict__ W,
              const float* __restrict__ bi, const float* __restrict__ bh,
              float* __restrict__ out)
{
  const int lane = threadIdx.x & 31;
  const int wid  = blockIdx.x * 8 + (threadIdx.x >> 5);
  const int nt   = wid & 15;                 // N tile (16 cols)
  const int mg   = wid >> 4;                 // group of 4 M tiles (64 rows)
  const int col  = nt * 16 + (lane & 15);    // output column (also B-matrix N)
  const int kh   = (lane < 16) ? 0 : 2;      // K sub-offset inside a 16x16x4 fragment
  const int arow = mg * 64 + (lane & 15);    // A row for sub-tile 0
  const int crow = (lane < 16) ? 0 : 8;      // C/D row offset

  const float* wp = W + (size_t)col * Hsz + kh;   // B[k][n] = W[n*H + k]
  const float* xp = X + (size_t)arow * Hsz + kh;

  v8f c0 = {}, c1 = {}, c2 = {}, c3 = {};
  #pragma unroll
  for (int k = 0; k < KT; ++k) {
    v2f b  = *(const v2f*)(wp + 4 * k);
    v2f a0 = *(const v2f*)(xp + 4 * k);
    v2f a1 = *(const v2f*)(xp + (size_t)16 * Hsz + 4 * k);
    v2f a2 = *(const v2f*)(xp + (size_t)32 * Hsz + 4 * k);
    v2f a3 = *(const v2f*)(xp + (size_t)48 * Hsz + 4 * k);
    c0 = __builtin_amdgcn_wmma_f32_16x16x4_f32(false, a0, false, b, (short)0, c0, false, false);
    c1 = __builtin_amdgcn_wmma_f32_16x16x4_f32(false, a1, false, b, (short)0, c1, false, false);
    c2 = __builtin_amdgcn_wmma_f32_16x16x4_f32(false, a2, false, b, (short)0, c2, false, false);
    c3 = __builtin_amdgcn_wmma_f32_16x16x4_f32(false, a3, false, b, (short)0, c3, false, false);
  }

  const float bias = bi[col] + bh[col];
  const size_t r0 = (size_t)mg * 64;
  #pragma unroll
  for (int i = 0; i < 8; ++i) {
    out[(r0 +  0 + crow + i) * Hsz + col] = c0[i] + bias;
    out[(r0 + 16 + crow + i) * Hsz + col] = c1[i] + bias;
    out[(r0 + 32 + crow + i) * Hsz + col] = c2[i] + bias;
    out[(r0 + 48 + crow + i) * Hsz + col] = c3[i] + bias;
  }
}

// --------------------------------------------------------------------------
// rnn_recur: h_t = tanh(pre_t + h_{t-1} @ Whh^T), h_{-1} = 0, over T steps.
// Grid = B/16 workgroups (batch rows are independent -> no inter-WG sync).
// Workgroup = 512 threads = 16 waves; wave j owns output columns 16j..16j+15.
// Whh^T column slab lives in registers (64 x v2f = 128 VGPRs) for all T.
// Hidden state [16 x 256] ping-pongs through LDS. pre/h may alias (in-place).
// --------------------------------------------------------------------------
__global__ __launch_bounds__(512)
void rnn_recur(const float* __restrict__ pre, float* __restrict__ hout,
               const float* __restrict__ Whh)
{
  const int lane = threadIdx.x & 31;
  const int wave = threadIdx.x >> 5;          // 0..15 -> N tile
  const int r0   = blockIdx.x * 16;           // batch rows owned by this WG
  const int col  = wave * 16 + (lane & 15);   // global hidden column
  const int kh   = (lane < 16) ? 0 : 2;
  const int crow = (lane < 16) ? 0 : 8;

  __shared__ float hbuf[2][16][HS];

  // Resident B fragments: bfr[k] = { Whh[col*H + 4k+kh], Whh[col*H + 4k+kh+1] }
  v2f bfr[KT];
  {
    const float* wp = Whh + (size_t)col * Hsz + kh;
    #pragma unroll
    for (int k = 0; k < KT; ++k) bfr[k] = *(const v2f*)(wp + 4 * k);
  }

  // h_{-1} = 0
  for (int i = threadIdx.x; i < 16 * HS; i += 512) (&hbuf[0][0][0])[i] = 0.0f;
  __syncthreads();

  // prefetch pre for t = 0 (8 elements: rows crow..crow+7 of this wave's slice)
  float pf[8];
  #pragma unroll
  for (int i = 0; i < 8; ++i)
    pf[i] = pre[((size_t)(r0 + crow + i) * Tsz + 0) * Hsz + col];

  int p = 0;
  for (int t = 0; t < Tsz; ++t) {
    v8f c0, c1 = {};
    #pragma unroll
    for (int i = 0; i < 8; ++i) c0[i] = pf[i];

    // prefetch next timestep's pre while WMMAs run
    if (t + 1 < Tsz) {
      #pragma unroll
      for (int i = 0; i < 8; ++i)
        pf[i] = pre[((size_t)(r0 + crow + i) * Tsz + (t + 1)) * Hsz + col];
    }

    // h_{t-1}[16x256] @ Whh^T slab: 64 x WMMA f32 16x16x4, two accum chains
    #pragma unroll
    for (int k = 0; k < KT; k += 2) {
      v2f a0 = *(const v2f*)(&hbuf[p][lane & 15][4 * k + kh]);
      c0 = __builtin_amdgcn_wmma_f32_16x16x4_f32(false, a0, false, bfr[k],
                                                 (short)0, c0, false, false);
      v2f a1 = *(const v2f*)(&hbuf[p][lane & 15][4 * (k + 1) + kh]);
      c1 = __builtin_amdgcn_wmma_f32_16x16x4_f32(false, a1, false, bfr[k + 1],
                                                 (short)0, c1, false, false);
    }

    // tanh, publish to LDS (other waves' next A) and to global output
    #pragma unroll
    for (int i = 0; i < 8; ++i) {
      const float h = tanhf(c0[i] + c1[i]);
      hbuf[1 - p][crow + i][col] = h;
      hout[((size_t)(r0 + crow + i) * Tsz + t) * Hsz + col] = h;
    }
    p = 1 - p;
    __syncthreads();
  }
}

// --------------------------------------------------------------------------
extern "C" void kernel_launch(void* const* d_in, const int* in_sizes, int n_in,
                              void* d_out, int out_size, void* d_ws, size_t ws_size,
                              hipStream_t stream)
{
  const float* x     = (const float*)d_in[0];
  const float* Wih0  = (const float*)d_in[1];
  const float* Whh0  = (const float*)d_in[2];
  const float* bih0  = (const float*)d_in[3];
  const float* bhh0  = (const float*)d_in[4];
  const float* Wih1  = (const float*)d_in[5];
  const float* Whh1  = (const float*)d_in[6];
  const float* bih1  = (const float*)d_in[7];
  const float* bhh1  = (const float*)d_in[8];
  float*       out   = (float*)d_out;
  float*       ws    = (float*)d_ws;     // one [B,T,H] f32 buffer (128 MB)

  // M = B*T = 131072 rows; 64 rows per wave-slab; 16 N tiles; 8 waves/block
  const int blocksA = (Bsz * Tsz / 64) * 16 / 8;   // 4096
  const int blocksR = Bsz / 16;                    // 4

  // layer 0
  pre_gemm <<<blocksA, 256, 0, stream>>>(x,  Wih0, bih0, bhh0, ws);
  rnn_recur<<<blocksR, 512, 0, stream>>>(ws, ws,   Whh0);      // in-place pre0 -> h0
  // layer 1
  pre_gemm <<<blocksA, 256, 0, stream>>>(ws, Wih1, bih1, bhh1, out);
  rnn_recur<<<blocksR, 512, 0, stream>>>(out, out, Whh1);      // in-place pre1 -> h1
}